// GeometrySeparationLoss_4312147165277
// MI455X (gfx1250) — compile-verified
//
#include <hip/hip_runtime.h>

typedef __attribute__((ext_vector_type(2))) float v2f;
typedef __attribute__((ext_vector_type(8))) float v8f;

#define D_DIM 768
#define R_DIM 128
#define B_DIM 512
#define C_NEW 10
#define C_OLD 100
#define CHUNK 128
#define CPAD  (R_DIM + 4)   // 132-float row stride: lane stride 4 banks -> conflict-free
#define MARGIN_F 5.0f
#define VAR_FLOOR_F 1e-4f
#define BIG_F 1e6f

// nan_to_num(nan=0, posinf=0, neginf=0): exponent-all-ones -> 0
__device__ __forceinline__ float nz(float x) {
    unsigned u = __float_as_uint(x);
    return ((u & 0x7F800000u) == 0x7F800000u) ? 0.0f : x;
}

__device__ __forceinline__ float half16_reduce(float v) {
    // sum across the 16 lanes of each wave half (masks keep bit4 fixed)
    v += __shfl_xor(v, 1);
    v += __shfl_xor(v, 2);
    v += __shfl_xor(v, 4);
    v += __shfl_xor(v, 8);
    return v;
}

// ---- Pass 0: clean features once (hoists nan_to_num out of GEMM hot loops)
// and initialize the old-class running min.
__global__ __launch_bounds__(256) void geom_clean_kernel(
    const float* __restrict__ feat, float* __restrict__ z_clean,
    unsigned* __restrict__ old_min_bits)
{
    const int i = blockIdx.x * blockDim.x + threadIdx.x;   // float4 index
    if (i < (B_DIM * D_DIM) / 4) {
        float4 v = ((const float4*)feat)[i];
        v.x = nz(v.x); v.y = nz(v.y); v.z = nz(v.z); v.w = nz(v.w);
        ((float4*)z_clean)[i] = v;
    }
    if (i < B_DIM) old_min_bits[i] = 0x7F800000u;          // +inf bit pattern
}

// ---- Pass 1: per-class geometry distance, one workgroup (8 waves) per class.
__global__ __launch_bounds__(256) void geom_dist_kernel(
    const float* __restrict__ z_clean,
    const float* __restrict__ cur_means, const float* __restrict__ cur_bases,
    const float* __restrict__ cur_vars,
    const float* __restrict__ old_means, const float* __restrict__ old_bases,
    const float* __restrict__ old_vars,
    float* __restrict__ dist_new, unsigned* __restrict__ old_min_bits)
{
    __shared__ float coeff_s[CHUNK][CPAD];
    __shared__ float parallel_s[CHUNK];

    const int cc = blockIdx.x;                 // 0..109
    const bool is_new = (cc < C_NEW);
    const int ci = is_new ? cc : cc - C_NEW;
    const float* mean = is_new ? cur_means + (size_t)ci * D_DIM
                               : old_means + (size_t)ci * D_DIM;
    const float* bas  = is_new ? cur_bases + (size_t)ci * D_DIM * R_DIM
                               : old_bases + (size_t)ci * D_DIM * R_DIM;
    const float* var  = is_new ? cur_vars + (size_t)ci * (R_DIM + 1)
                               : old_vars + (size_t)ci * (R_DIM + 1);
    const float inv_res = 1.0f / fmaxf(var[R_DIM], VAR_FLOOR_F);

    const int wave = threadIdx.x >> 5;
    const int lane = threadIdx.x & 31;
    const int hl = lane >> 4;     // wave half
    const int ll = lane & 15;
    const int kb = hl * 2;        // K sub-offset for A/B operands (f32 16x16x4 layout)

    for (int b0 = 0; b0 < B_DIM; b0 += CHUNK) {
        // ================= Stage A: coeff = (z - mean) @ bases =================
        // M=128, N=128, K=768. Wave w owns rows [w*16, w*16+16) and ALL 8 N-tiles:
        // k-outer / nt-inner with 8 independent accumulators -> A built once per
        // K-step, 8 back-to-back independent v_wmma ops per A fetch.
        const float* zrow = z_clean + (size_t)(b0 + wave * 16 + ll) * D_DIM;

        v8f acc[8];
#pragma unroll
        for (int nt = 0; nt < 8; ++nt)
            acc[nt] = (v8f){0.f, 0.f, 0.f, 0.f, 0.f, 0.f, 0.f, 0.f};

#pragma unroll 2
        for (int k0 = 0; k0 < D_DIM; k0 += 4) {
            const int k = k0 + kb;
            float2 zf = *(const float2*)(zrow + k);
            float2 mf = *(const float2*)(mean + k);
            v2f a; a.x = zf.x - mf.x; a.y = zf.y - mf.y;
            const float* bk0 = bas + (size_t)k * R_DIM + ll;
            const float* bk1 = bk0 + R_DIM;
#pragma unroll
            for (int nt = 0; nt < 8; ++nt) {
                v2f b; b.x = bk0[nt * 16]; b.y = bk1[nt * 16];
                acc[nt] = __builtin_amdgcn_wmma_f32_16x16x4_f32(
                    false, a, false, b, (short)0, acc[nt], false, false);
            }
        }

        // Epilogue: spill coeff tiles to LDS + parallel term sum(coeff^2/eig)
        float pacc[8];
#pragma unroll
        for (int i = 0; i < 8; ++i) pacc[i] = 0.0f;
#pragma unroll
        for (int nt = 0; nt < 8; ++nt) {
            const int n = nt * 16 + ll;
            const float inv_eig = 1.0f / fmaxf(var[n], VAR_FLOOR_F);
#pragma unroll
            for (int i = 0; i < 8; ++i) {
                const int lr = wave * 16 + i + 8 * hl;   // C layout: VGPR i -> M = i + 8*half
                coeff_s[lr][n] = acc[nt][i];
                pacc[i] += acc[nt][i] * acc[nt][i] * inv_eig;
            }
        }
#pragma unroll
        for (int i = 0; i < 8; ++i) {
            float v = half16_reduce(pacc[i]);
            if (ll == 0) parallel_s[wave * 16 + i + 8 * hl] = v;
        }
        __syncthreads();

        // ================= Stage B: residual = ||diff - coeff @ basesT||^2 =====
        // M=128 rows, N=768 d-cols, K=128. A operand (this wave's coeff band) is
        // preloaded into 32 v2f registers; inner loop is 1 float2 B load + 1 wmma.
        const int arow = wave * 16 + ll;
        v2f areg[32];
#pragma unroll
        for (int kk = 0; kk < 32; ++kk) {
            float2 cf = *(const float2*)&coeff_s[arow][kk * 4 + kb];
            areg[kk].x = cf.x; areg[kk].y = cf.y;
        }

        float racc[8];
#pragma unroll
        for (int i = 0; i < 8; ++i) racc[i] = 0.0f;

        for (int dt = 0; dt < D_DIM / 16; ++dt) {
            const int d = dt * 16 + ll;
            const float* brow = bas + (size_t)d * R_DIM + kb;   // basesT row-contig
            v8f r = (v8f){0.f, 0.f, 0.f, 0.f, 0.f, 0.f, 0.f, 0.f};
#pragma unroll
            for (int kk = 0; kk < 32; ++kk) {
                float2 bf = *(const float2*)(brow + kk * 4);
                v2f b; b.x = bf.x; b.y = bf.y;
                r = __builtin_amdgcn_wmma_f32_16x16x4_f32(
                    false, areg[kk], false, b, (short)0, r, false, false);
            }
            const float mf = mean[d];
#pragma unroll
            for (int i = 0; i < 8; ++i) {
                const int grow = b0 + wave * 16 + i + 8 * hl;
                float diff = z_clean[(size_t)grow * D_DIM + d] - mf;
                float e = diff - r[i];
                racc[i] += e * e;
            }
        }
#pragma unroll
        for (int i = 0; i < 8; ++i) {
            float v = half16_reduce(racc[i]);
            if (ll == 0) {
                const int lrow = wave * 16 + i + 8 * hl;
                float dist = (parallel_s[lrow] + v * inv_res) * (1.0f / (float)D_DIM);
                unsigned u = __float_as_uint(dist);
                if ((u & 0x7F800000u) == 0x7F800000u) dist = BIG_F;  // nan_to_num -> BIG
                const int grow = b0 + lrow;
                if (is_new) {
                    dist_new[grow * C_NEW + ci] = dist;
                } else {
                    // dist >= 0 so uint bit-pattern ordering == float ordering
                    atomicMin(old_min_bits + grow, __float_as_uint(dist));
                }
            }
        }
        __syncthreads();
    }
}

__device__ float block_reduce256(float v, float* red) {
    const int t = threadIdx.x;
    red[t] = v;
    __syncthreads();
    for (int s = 128; s > 0; s >>= 1) {
        if (t < s) red[t] += red[t + s];
        __syncthreads();
    }
    float r = red[0];
    __syncthreads();
    return r;
}

// ---- Pass 2: masked per-class means + hinge -> 3 scalars
__global__ __launch_bounds__(256) void geom_finalize_kernel(
    const float* __restrict__ dist_new, const unsigned* __restrict__ old_min_bits,
    const int* __restrict__ labels, const int* __restrict__ ids,
    float* __restrict__ out)
{
    __shared__ float red[256];
    const int t = threadIdx.x;
    float total_a = 0.f, own_a = 0.f, old_a = 0.f, nvalid = 0.f;

    for (int k = 0; k < C_NEW; ++k) {
        int id = ids[k];
        int col = id < 0 ? 0 : (id >= C_NEW ? C_NEW - 1 : id);
        float cnt = 0.f, owns = 0.f, olds = 0.f, hs = 0.f;
        for (int b = t; b < B_DIM; b += 256) {
            if (labels[b] == id) {
                float ow = dist_new[b * C_NEW + col];
                float om = __uint_as_float(old_min_bits[b]);
                cnt += 1.f;
                owns += ow;
                olds += om;
                float h = MARGIN_F + ow - om;
                hs += (h > 0.f) ? h : 0.f;
            }
        }
        cnt  = block_reduce256(cnt,  red);
        owns = block_reduce256(owns, red);
        olds = block_reduce256(olds, red);
        hs   = block_reduce256(hs,   red);
        if (t == 0 && cnt > 0.f) {
            total_a += hs / cnt;
            own_a   += owns / cnt;
            old_a   += olds / cnt;
            nvalid  += 1.f;
        }
    }
    if (t == 0) {
        float nv = fmaxf(nvalid, 1.f);
        out[0] = total_a / nv;
        out[1] = own_a / nv;
        out[2] = old_a / nv;
    }
}

extern "C" void kernel_launch(void* const* d_in, const int* in_sizes, int n_in,
                              void* d_out, int out_size, void* d_ws, size_t ws_size,
                              hipStream_t stream) {
    const float* feat      = (const float*)d_in[0];
    const int*   labels    = (const int*)d_in[1];
    const int*   ids       = (const int*)d_in[2];
    const float* cur_means = (const float*)d_in[3];
    const float* cur_bases = (const float*)d_in[4];
    const float* cur_vars  = (const float*)d_in[5];
    const float* old_means = (const float*)d_in[6];
    const float* old_bases = (const float*)d_in[7];
    const float* old_vars  = (const float*)d_in[8];

    // workspace layout
    float*    z_clean      = (float*)d_ws;                         // 512*768 f32
    float*    dist_new     = z_clean + (size_t)B_DIM * D_DIM;      // 512*10  f32
    unsigned* old_min_bits = (unsigned*)(dist_new + B_DIM * C_NEW); // 512 u32

    geom_clean_kernel<<<(B_DIM * D_DIM / 4 + 255) / 256, 256, 0, stream>>>(
        feat, z_clean, old_min_bits);
    geom_dist_kernel<<<C_NEW + C_OLD, 256, 0, stream>>>(
        z_clean, cur_means, cur_bases, cur_vars,
        old_means, old_bases, old_vars, dist_new, old_min_bits);
    geom_finalize_kernel<<<1, 256, 0, stream>>>(dist_new, old_min_bits, labels, ids,
                                                (float*)d_out);
}